// VectorQuantizer_37658273251489
// MI455X (gfx1250) — compile-verified
//
#include <hip/hip_runtime.h>
#include <hip/hip_bf16.h>

typedef float v2f __attribute__((ext_vector_type(2)));
typedef float v8f __attribute__((ext_vector_type(8)));

#define N_ROWS 65536
#define DIM    256
#define KCODE  1024

// output layout (floats, concatenated in reference return order)
#define Q_OFF    0                 // quantized_st: 64*1024*256
#define VQ_OFF   16777216          // vq_loss scalar
#define PP_OFF   16777217          // perplexity scalar
#define IND_OFF  16777218          // encoding_inds: 65536
#define DIST_OFF 16842754          // dist: 65536*1024

#define TILE_M 128
#define TILE_N 128
#define TILE_K 32
#define LDS_STRIDE 36              // pad: conflict-free + 16B aligned float4 stores

// ---------------------------------------------------------------- row ||.||^2
__global__ __launch_bounds__(256) void vq_row_sumsq(const float* __restrict__ A,
                                                    float* __restrict__ out,
                                                    int rows) {
    int wid  = (blockIdx.x * 256 + threadIdx.x) >> 5;
    int lane = threadIdx.x & 31;
    if (wid >= rows) return;                 // launches are exact multiples of 8 waves
    const float* r = A + (size_t)wid * DIM;
    float s = 0.f;
#pragma unroll
    for (int j = 0; j < 2; ++j) {
        float4 v = *(const float4*)(r + lane * 4 + j * 128);
        s += v.x * v.x + v.y * v.y + v.z * v.z + v.w * v.w;
    }
    for (int off = 16; off; off >>= 1) s += __shfl_xor(s, off);
    if (lane == 0) out[wid] = s;
}

// ---------------------------------------------------------------- WMMA GEMM
// dist[m][n] = xsq[m] + esq[n] - 2 * dot(X[m,:], W[n,:])
__global__ __launch_bounds__(256) void vq_gemm_dist(const float* __restrict__ X,
                                                    const float* __restrict__ W,
                                                    const float* __restrict__ xsq,
                                                    const float* __restrict__ esq,
                                                    float* __restrict__ dist) {
    __shared__ float Xs[TILE_M * LDS_STRIDE];
    __shared__ float Ws[TILE_N * LDS_STRIDE];

    const int tid    = threadIdx.x;
    const int lane   = tid & 31;
    const int wid    = tid >> 5;
    const int wave_m = wid & 1;        // 0..1 -> 64 rows each
    const int wave_n = wid >> 1;       // 0..3 -> 32 cols each
    const int gm0    = blockIdx.y * TILE_M;
    const int gn0    = blockIdx.x * TILE_N;
    const int l16    = lane & 15;
    const int koff   = (lane >> 4) << 1;   // A/B frag: K offset 0 or 2
    const int mrow8  = (lane >> 4) << 3;   // C/D frag: +8 rows for lanes 16..31

    v8f acc[4][2] = {};

    for (int d0 = 0; d0 < DIM; d0 += TILE_K) {
        // stage 128x32 tiles of X and W into LDS (4 float4 per thread per tile)
#pragma unroll
        for (int j = 0; j < 4; ++j) {
            int idx = tid + 256 * j;           // 0..1023
            int row = idx >> 3;
            int c4  = (idx & 7) << 2;
            *(float4*)(Xs + row * LDS_STRIDE + c4) =
                *(const float4*)(X + (size_t)(gm0 + row) * DIM + d0 + c4);
            *(float4*)(Ws + row * LDS_STRIDE + c4) =
                *(const float4*)(W + (size_t)(gn0 + row) * DIM + d0 + c4);
        }
        __syncthreads();

#pragma unroll
        for (int dd = 0; dd < TILE_K; dd += 4) {
            v2f a[4], b[2];
#pragma unroll
            for (int mi = 0; mi < 4; ++mi)
                a[mi] = *(const v2f*)(Xs + (wave_m * 64 + mi * 16 + l16) * LDS_STRIDE + dd + koff);
#pragma unroll
            for (int ni = 0; ni < 2; ++ni)
                b[ni] = *(const v2f*)(Ws + (wave_n * 32 + ni * 16 + l16) * LDS_STRIDE + dd + koff);
#pragma unroll
            for (int mi = 0; mi < 4; ++mi)
#pragma unroll
                for (int ni = 0; ni < 2; ++ni)
                    acc[mi][ni] = __builtin_amdgcn_wmma_f32_16x16x4_f32(
                        false, a[mi], false, b[ni], (short)0, acc[mi][ni], false, false);
        }
        __syncthreads();
    }

    // epilogue: dist = xsq + esq - 2*dot
#pragma unroll
    for (int mi = 0; mi < 4; ++mi) {
#pragma unroll
        for (int ni = 0; ni < 2; ++ni) {
#pragma unroll
            for (int v = 0; v < 8; ++v) {
                int m = gm0 + wave_m * 64 + mi * 16 + mrow8 + v;
                int n = gn0 + wave_n * 32 + ni * 16 + l16;
                dist[(size_t)m * KCODE + n] = xsq[m] + esq[n] - 2.0f * acc[mi][ni][v];
            }
        }
    }
}

// ------------------------------------------------- argmin + gather + hist
__global__ __launch_bounds__(256) void vq_argmin_gather(const float* __restrict__ dist,
                                                        const float* __restrict__ X,
                                                        const float* __restrict__ W,
                                                        float* __restrict__ qout,
                                                        float* __restrict__ indout,
                                                        float* __restrict__ rowsq,
                                                        unsigned int* __restrict__ hist) {
    int row  = (blockIdx.x * 256 + threadIdx.x) >> 5;
    int lane = threadIdx.x & 31;
    const float* dr = dist + (size_t)row * KCODE;

    float bv = 3.402823466e38f;
    int   bk = 0;
#pragma unroll
    for (int i = 0; i < 8; ++i) {
        int base = i * 128 + lane * 4;
        float4 v = *(const float4*)(dr + base);
        if (v.x < bv) { bv = v.x; bk = base;     }
        if (v.y < bv) { bv = v.y; bk = base + 1; }
        if (v.z < bv) { bv = v.z; bk = base + 2; }
        if (v.w < bv) { bv = v.w; bk = base + 3; }
    }
    // wave32 butterfly min with first-index tie-break (matches jnp.argmin)
    for (int off = 16; off; off >>= 1) {
        float ov = __shfl_xor(bv, off);
        int   ok = __shfl_xor(bk, off);
        if (ov < bv || (ov == bv && ok < bk)) { bv = ov; bk = ok; }
    }

    const float* wrow = W + (size_t)bk * DIM;
    const float* xrow = X + (size_t)row * DIM;
    float*       qrow = qout + (size_t)row * DIM;
    float s = 0.f;
#pragma unroll
    for (int j = 0; j < 2; ++j) {
        int c = lane * 4 + j * 128;
        float4 q = *(const float4*)(wrow + c);
        float4 x = *(const float4*)(xrow + c);
        *(float4*)(qrow + c) = q;
        float dx = q.x - x.x, dy = q.y - x.y, dz = q.z - x.z, dw = q.w - x.w;
        s += dx * dx + dy * dy + dz * dz + dw * dw;
    }
    for (int off = 16; off; off >>= 1) s += __shfl_xor(s, off);
    if (lane == 0) {
        rowsq[row]  = s;                       // deterministic: reduced later in fixed order
        indout[row] = (float)bk;
        atomicAdd(&hist[bk], 1u);              // integer atomic: deterministic
    }
}

// ---------------------------------------------------------------- finalize
__global__ __launch_bounds__(1024) void vq_finalize(const float* __restrict__ rowsq,
                                                    const unsigned int* __restrict__ hist,
                                                    float* __restrict__ out_vq,
                                                    float* __restrict__ out_pp) {
    __shared__ float sh_sq[1024];
    __shared__ float sh_en[1024];
    int t = threadIdx.x;
    float s = 0.f;
    for (int i = t; i < N_ROWS; i += 1024) s += rowsq[i];   // fixed order per thread
    float p = (float)hist[t] * (1.0f / (float)N_ROWS);
    sh_sq[t] = s;
    sh_en[t] = p * logf(p + 1e-10f);
    __syncthreads();
    for (int off = 512; off; off >>= 1) {
        if (t < off) { sh_sq[t] += sh_sq[t + off]; sh_en[t] += sh_en[t + off]; }
        __syncthreads();
    }
    if (t == 0) {
        float msq = sh_sq[0] / ((float)N_ROWS * (float)DIM);
        out_vq[0] = msq * (0.25f + 1.0f);      // commitment*BETA + DELTA*embedding (equal fwd)
        out_pp[0] = expf(-sh_en[0]);
    }
}

__global__ void vq_zero_hist(unsigned int* hist) {
    int t = blockIdx.x * blockDim.x + threadIdx.x;
    if (t < KCODE) hist[t] = 0u;
}

extern "C" void kernel_launch(void* const* d_in, const int* in_sizes, int n_in,
                              void* d_out, int out_size, void* d_ws, size_t ws_size,
                              hipStream_t stream) {
    const float* X = (const float*)d_in[0];     // latents_mean [64,1024,256] fp32
    const float* W = (const float*)d_in[1];     // embedding_weight [1024,256] fp32
    float* out  = (float*)d_out;
    float* q    = out + Q_OFF;
    float* vq   = out + VQ_OFF;
    float* pp   = out + PP_OFF;
    float* inds = out + IND_OFF;
    float* dist = out + DIST_OFF;

    float* ws   = (float*)d_ws;
    float* xsq  = ws;                               // 65536
    float* esq  = ws + 65536;                       // 1024
    unsigned int* hist = (unsigned int*)(ws + 66560); // 1024
    float* rowsq = ws + 67584;                      // 65536

    vq_zero_hist<<<4, 256, 0, stream>>>(hist);
    vq_row_sumsq<<<N_ROWS / 8, 256, 0, stream>>>(X, xsq, N_ROWS);
    vq_row_sumsq<<<KCODE / 8, 256, 0, stream>>>(W, esq, KCODE);
    vq_gemm_dist<<<dim3(KCODE / TILE_N, N_ROWS / TILE_M), 256, 0, stream>>>(X, W, xsq, esq, dist);
    vq_argmin_gather<<<N_ROWS / 8, 256, 0, stream>>>(dist, X, W, q, inds, rowsq, hist);
    vq_finalize<<<1, 1024, 0, stream>>>(rowsq, hist, vq, pp);
}